// RocAucLossCE_13950053777743
// MI455X (gfx1250) — compile-verified
//
#include <hip/hip_runtime.h>
#include <hip/hip_bf16.h>

typedef float v2f __attribute__((ext_vector_type(2)));
typedef float v8f __attribute__((ext_vector_type(8)));

#define GAMMA 0.2f
#define TI    1024   // i-span per block (4 per thread)
#define TJ    1024   // j-span per block (staged in LDS)
#define BT    256    // threads per block (8 wave32s)

// ---------------------------------------------------------------------------
// Kernel 1: per-sample softmax prep.
//   s = 1 - softmax(x)[0];  A[i] = neg-row value (else -1e30 sentinel)
//                           B[j] = pos-col value (else +1e30 sentinel)
// Sentinels make gamma + A[i] - B[j] hugely negative for any invalid pair,
// so relu() kills it with no branches (finite: 2e30 << FLT_MAX).
// ---------------------------------------------------------------------------
__global__ void __launch_bounds__(256)
prep_kernel(const float4* __restrict__ yp, const int* __restrict__ yt,
            float* __restrict__ Aneg, float* __restrict__ Bpos, int n) {
  int i = blockIdx.x * blockDim.x + threadIdx.x;
  if (i >= n) return;
  float4 x = yp[i];
  float m  = fmaxf(fmaxf(x.x, x.y), fmaxf(x.z, x.w));
  float e0 = expf(x.x - m);
  float e1 = expf(x.y - m);
  float e2 = expf(x.z - m);
  float e3 = expf(x.w - m);
  float s  = 1.0f - e0 / (((e0 + e1) + (e2 + e3)));
  int lab  = yt[i];
  Aneg[i] = (lab == 0) ? s : -1e30f;   // row (negative) operand
  Bpos[i] = (lab != 0) ? s :  1e30f;   // col (positive) operand
}

// one relu-cube pairwise update: acc += relu(a - b)^3   (a already has +gamma)
#define UPD(ak, bv_, acck)                                   \
  { float d_ = (ak) - (bv_);                                 \
    float t_ = fmaxf(d_, 0.0f);                              \
    (acck)   = fmaf(t_ * t_, t_, (acck)); }

// ---------------------------------------------------------------------------
// Kernel 2: tiled pairwise accumulation. Each block covers a 1024x1024 tile.
// Each thread keeps 4 row operands in registers and streams the column tile
// from LDS as float4 broadcasts (16 fused updates per ds_load_b128).
// Wave reduction is done with V_WMMA_F32_16X16X4_F32 (B = ones).
// ---------------------------------------------------------------------------
__global__ void __launch_bounds__(BT)
pair_kernel(const float* __restrict__ Aneg, const float* __restrict__ Bpos,
            float* __restrict__ partials) {
  __shared__ float lb[TJ];
  __shared__ float wsum[16];

  const int tid = threadIdx.x;
  const int i0  = blockIdx.x * TI;
  const int j0  = blockIdx.y * TJ;

  // stage the column tile: 256 threads x float4 = 1024 floats (coalesced)
  ((float4*)lb)[tid] = ((const float4*)(Bpos + j0))[tid];

  // 4 row operands per thread, pre-biased by gamma
  float a0 = GAMMA + Aneg[i0 + 0 * BT + tid];
  float a1 = GAMMA + Aneg[i0 + 1 * BT + tid];
  float a2 = GAMMA + Aneg[i0 + 2 * BT + tid];
  float a3 = GAMMA + Aneg[i0 + 3 * BT + tid];

  __syncthreads();

  float acc0 = 0.f, acc1 = 0.f, acc2 = 0.f, acc3 = 0.f;
  const float4* lb4 = (const float4*)lb;
#pragma unroll 2
  for (int jj = 0; jj < TJ / 4; ++jj) {
    float4 b = lb4[jj];                 // broadcast read: conflict-free
    UPD(a0, b.x, acc0) UPD(a1, b.x, acc1) UPD(a2, b.x, acc2) UPD(a3, b.x, acc3)
    UPD(a0, b.y, acc0) UPD(a1, b.y, acc1) UPD(a2, b.y, acc2) UPD(a3, b.y, acc3)
    UPD(a0, b.z, acc0) UPD(a1, b.z, acc1) UPD(a2, b.z, acc2) UPD(a3, b.z, acc3)
    UPD(a0, b.w, acc0) UPD(a1, b.w, acc1) UPD(a2, b.w, acc2) UPD(a3, b.w, acc3)
  }
  float acc = (acc0 + acc1) + (acc2 + acc3);

  // ---- wave32 reduction via WMMA: D = A(acc,0) x B(ones) + 0 --------------
  // A 16x4 layout: lane L<16 -> A[L][0]=acc; lane L>=16 -> A[L-16][2]=acc.
  // With B == ones, D[m][n] = acc_m + acc_{m+16}; summing the 8 D VGPRs in a
  // lane yields sum over rows 0..7 (lanes 0-15) / rows 8..15 (lanes 16-31).
  v2f av; av.x = acc;  av.y = 0.0f;
  v2f bv; bv.x = 1.0f; bv.y = 1.0f;
  v8f cv = {};
  v8f dv = __builtin_amdgcn_wmma_f32_16x16x4_f32(
      /*neg_a=*/false, av, /*neg_b=*/false, bv,
      /*c_mod=*/(short)0, cv, /*reuse_a=*/false, /*reuse_b=*/false);
  float r = ((dv[0] + dv[1]) + (dv[2] + dv[3])) +
            ((dv[4] + dv[5]) + (dv[6] + dv[7]));

  const int lane = tid & 31;
  const int wave = tid >> 5;
  if (lane == 0)  wsum[wave * 2 + 0] = r;   // rows 0..7  half-sum
  if (lane == 16) wsum[wave * 2 + 1] = r;   // rows 8..15 half-sum
  __syncthreads();

  if (tid == 0) {
    float t = 0.0f;
#pragma unroll
    for (int w = 0; w < 16; ++w) t += wsum[w];   // fixed order: deterministic
    partials[blockIdx.y * gridDim.x + blockIdx.x] = t;
  }
}

// ---------------------------------------------------------------------------
// Kernel 3: fold block partials in a fixed serial order (deterministic).
// ---------------------------------------------------------------------------
__global__ void final_kernel(const float* __restrict__ partials,
                             float* __restrict__ out, int nparts) {
  if (threadIdx.x == 0 && blockIdx.x == 0) {
    float t = 0.0f;
    for (int i = 0; i < nparts; ++i) t += partials[i];
    out[0] = t;
  }
}

// ---------------------------------------------------------------------------
extern "C" void kernel_launch(void* const* d_in, const int* in_sizes, int n_in,
                              void* d_out, int out_size, void* d_ws, size_t ws_size,
                              hipStream_t stream) {
  const float* yp = (const float*)d_in[0];   // [N,4] f32 logits
  const int*   yt = (const int*)d_in[1];     // [N] labels 0..3
  const int n = in_sizes[1];                 // 16384

  float* Aneg     = (float*)d_ws;            // n floats
  float* Bpos     = Aneg + n;                // n floats
  float* partials = Bpos + n;                // (n/TI)*(n/TJ) floats

  const int gi = n / TI;                     // 16
  const int gj = n / TJ;                     // 16

  prep_kernel<<<(n + 255) / 256, 256, 0, stream>>>(
      (const float4*)yp, yt, Aneg, Bpos, n);
  pair_kernel<<<dim3(gi, gj), BT, 0, stream>>>(Aneg, Bpos, partials);
  final_kernel<<<1, 32, 0, stream>>>(partials, (float*)d_out, gi * gj);
}